// ImageRotator_1305670058687
// MI455X (gfx1250) — compile-verified
//
#include <hip/hip_runtime.h>
#include <stdint.h>

// ---------------------------------------------------------------------------
// Batched image rotation (32 images x 8 angles, 512x512, bilinear).
// Strategy (MI455X / gfx1250):
//   - workgroup = (angle, 64x64 output tile); loop over 32 images inside.
//   - Tensor Data Mover DMAs the rotated-source bounding box (<=92x92 f32,
//     <=34KB) into LDS, double-buffered, tracked with TENSORcnt.
//   - per-pixel rotation math hoisted out of the image loop (reused 32x).
//   - output (256MB stream) written with non-temporal B128 stores so it does
//     not thrash the 192MB L2 that holds the 32MB input.
//   - hardware v_sin/v_cos (angles are small; no Payne-Hanek needed).
// ---------------------------------------------------------------------------

typedef __attribute__((ext_vector_type(4))) unsigned int v4u;
typedef __attribute__((ext_vector_type(8))) int          v8i;
typedef __attribute__((ext_vector_type(4))) int          v4i;
typedef __attribute__((ext_vector_type(4))) float        v4f;

constexpr int W    = 512;
constexpr int H    = 512;
constexpr int NIMG = 32;
constexpr int NANG = 8;
constexpr int TS   = 64;               // output tile edge
constexpr int BMAX = 96;               // max bbox edge (63*sqrt(2)+3 < 96)
constexpr int BUFE = BMAX * BMAX;      // LDS floats per buffer

#if defined(__has_builtin)
#if __has_builtin(__builtin_amdgcn_tensor_load_to_lds) && \
    __has_builtin(__builtin_amdgcn_s_wait_tensorcnt)
#define HAVE_TDM 1
#endif
#endif
#ifndef HAVE_TDM
#define HAVE_TDM 0
#endif

__device__ __forceinline__ int imin(int a, int b) { return a < b ? a : b; }
__device__ __forceinline__ int imax(int a, int b) { return a > b ? a : b; }

#if HAVE_TDM
// Issue a TDM 2D tile load: bw x bh f32 tile from a 512-stride tensor -> LDS.
// D# packing per CDNA5 ISA ch.8 (groups 0/1; groups 2/3 zero for 2D tiles).
__device__ __forceinline__ void tdm_load_tile(const float* gsrc,
                                              unsigned lds_byte_off,
                                              int bw, int bh) {
    uint64_t ga = (uint64_t)(uintptr_t)gsrc;
    v4u g0;
    g0[0] = 1u;                                   // count=1, user descriptor
    g0[1] = lds_byte_off;                         // lds_addr [63:32]
    g0[2] = (unsigned)ga;                         // global_addr [95:64]
    g0[3] = ((unsigned)(ga >> 32) & 0x1FFFFFFu)   // global_addr [120:96]
          | (2u << 30);                           // type=2 ("image") [127:126]
    v8i g1;
    g1[0] = (2 << 16);                            // data_size=2 (4 bytes)
    g1[1] = (int)(((unsigned)W & 0xFFFFu) << 16); // tensor_dim0 lo16 @ [63:48]
    g1[2] = (int)((((unsigned)W >> 16) & 0xFFFFu) // tensor_dim0 hi16
          | (((unsigned)H & 0xFFFFu) << 16));     // tensor_dim1 lo16
    g1[3] = (int)((((unsigned)H >> 16) & 0xFFFFu) // tensor_dim1 hi16
          | (((unsigned)bw & 0xFFFFu) << 16));    // tile_dim0
    g1[4] = (int)((unsigned)bh & 0xFFFFu);        // tile_dim1 (tile_dim2=0)
    g1[5] = W;                                    // tensor_dim0_stride lo32
    g1[6] = 0;                                    // stride hi16 | dim1_stride lo
    g1[7] = 0;                                    // dim1_stride hi (unused, 2D)
    v4i g2 = {0, 0, 0, 0};
    v4i g3 = {0, 0, 0, 0};
#if __clang_major__ >= 23
    v8i g4 = {0, 0, 0, 0, 0, 0, 0, 0};
    __builtin_amdgcn_tensor_load_to_lds(g0, g1, g2, g3, g4, 0);
#else
    __builtin_amdgcn_tensor_load_to_lds(g0, g1, g2, g3, 0);
#endif
}
#endif // HAVE_TDM

__global__ __launch_bounds__(256) void
rotate_batch_kernel(const float* __restrict__ img,
                    const float* __restrict__ angles,
                    float* __restrict__ out) {
    __shared__ float smem[2 * BUFE];

    const int tx0 = blockIdx.x * TS;
    const int ty0 = blockIdx.y * TS;
    const int ang = blockIdx.z;
    const int lane = threadIdx.x;
    // SGPR-resident wave id -> guarantees a scalar branch around the TDM
    // issue (TDM ignores EXEC, so EXEC-masking alone would not gate it).
    const int wave = __builtin_amdgcn_readfirstlane((int)(threadIdx.x >> 5));

    const float a  = angles[ang] * 0.017453292519943295f; // deg2rad, < 2*pi
    const float ca = __cosf(a);   // v_cos_f32: no huge-arg reduction needed
    const float sa = __sinf(a);   // v_sin_f32
    const float cx = (W - 1) * 0.5f;
    const float cy = (H - 1) * 0.5f;

    // --- source bounding box of this output tile (linear map: extremes at
    //     the 4 tile corners), clamped into the image ---
    float minsx = 1e30f, maxsx = -1e30f, minsy = 1e30f, maxsy = -1e30f;
#pragma unroll
    for (int i = 0; i < 2; ++i) {
#pragma unroll
        for (int j = 0; j < 2; ++j) {
            float fx = (float)(tx0 + i * (TS - 1)) - cx;
            float fy = (float)(ty0 + j * (TS - 1)) - cy;
            float sxc = ca * fx - sa * fy + cx;
            float syc = sa * fx + ca * fy + cy;
            minsx = fminf(minsx, sxc); maxsx = fmaxf(maxsx, sxc);
            minsy = fminf(minsy, syc); maxsy = fmaxf(maxsy, syc);
        }
    }
    int bx0 = imin(W - 1, imax(0, (int)floorf(minsx)));
    int bx1 = imax(bx0, imin(W - 1, (int)floorf(maxsx) + 1));
    int by0 = imin(H - 1, imax(0, (int)floorf(minsy)));
    int by1 = imax(by0, imin(H - 1, (int)floorf(maxsy) + 1));
    const int bw = bx1 - bx0 + 1;  // <= 92 < BMAX
    const int bh = by1 - by0 + 1;

    const unsigned base_lds = (unsigned)(uintptr_t)(void*)smem;

#if HAVE_TDM
    // Prefetch image 0 into buffer 0 while we do the per-tile precompute.
    if (wave == 0) {
        tdm_load_tile(img + (size_t)by0 * W + bx0, base_lds, bw, bh);
    }
#endif

    // --- per-pixel sampling state, computed ONCE and reused for 32 images.
    // Thread t owns 4 row-groups x 4 consecutive columns = 16 pixels.
    int   o00[16], o01[16], o10[16], o11[16];   // LDS float offsets (clamped)
    float ax0[16], ax1[16], ay0[16], ay1[16];   // per-axis weights, OOB-masked
#pragma unroll
    for (int g = 0; g < 4; ++g) {
        const int ly = (lane >> 4) + 16 * g;
#pragma unroll
        for (int v = 0; v < 4; ++v) {
            const int idx = g * 4 + v;
            const int lx  = (lane & 15) * 4 + v;
            float fx = (float)(tx0 + lx) - cx;
            float fy = (float)(ty0 + ly) - cy;
            float sx = ca * fx - sa * fy + cx;
            float sy = sa * fx + ca * fy + cy;
            float x0f = floorf(sx), y0f = floorf(sy);
            float wx = sx - x0f,    wy = sy - y0f;
            int x0 = (int)x0f, y0 = (int)y0f;
            int x1 = x0 + 1,   y1 = y0 + 1;
            bool vx0 = (x0 >= 0) && (x0 < W);
            bool vx1 = (x1 >= 0) && (x1 < W);
            bool vy0 = (y0 >= 0) && (y0 < H);
            bool vy1 = (y1 >= 0) && (y1 < H);
            ax0[idx] = vx0 ? (1.0f - wx) : 0.0f;
            ax1[idx] = vx1 ? wx          : 0.0f;
            ay0[idx] = vy0 ? (1.0f - wy) : 0.0f;
            ay1[idx] = vy1 ? wy          : 0.0f;
            // valid taps always land in the bbox; clamp only protects the
            // fully-masked (weight==0) taps.
            int xr0 = imin(imax(x0 - bx0, 0), bw - 1);
            int xr1 = imin(imax(x1 - bx0, 0), bw - 1);
            int yr0 = imin(imax(y0 - by0, 0), bh - 1);
            int yr1 = imin(imax(y1 - by0, 0), bh - 1);
            o00[idx] = yr0 * bw + xr0;
            o01[idx] = yr0 * bw + xr1;
            o10[idx] = yr1 * bw + xr0;
            o11[idx] = yr1 * bw + xr1;
        }
    }

#if !HAVE_TDM
    // Fallback staging (no TDM builtin): cooperative copy, single buffer.
    for (int i = 0; i < NIMG; ++i) {
        __syncthreads();
        const float* src = img + (size_t)i * W * H + (size_t)by0 * W + bx0;
        for (int t = threadIdx.x; t < bw * bh; t += 256)
            smem[t] = src[(t / bw) * W + (t % bw)];
        __syncthreads();
        const float* sb = smem;
        float* op = out + ((size_t)i * NANG + ang) * (size_t)W * H;
#pragma unroll
        for (int g = 0; g < 4; ++g) {
            const int ly = (lane >> 4) + 16 * g;
            float r0, r1, r2, r3;
            float* rp[4] = {&r0, &r1, &r2, &r3};
#pragma unroll
            for (int v = 0; v < 4; ++v) {
                const int idx = g * 4 + v;
                float v00 = sb[o00[idx]], v01 = sb[o01[idx]];
                float v10 = sb[o10[idx]], v11 = sb[o11[idx]];
                *rp[v] = ay0[idx] * (ax0[idx] * v00 + ax1[idx] * v01)
                       + ay1[idx] * (ax0[idx] * v10 + ax1[idx] * v11);
            }
            v4f val = {r0, r1, r2, r3};
            float* dst = op + (size_t)(ty0 + ly) * W + tx0 + (lane & 15) * 4;
            __builtin_nontemporal_store(val, (v4f*)dst);
        }
    }
#else
    // --- TDM double-buffered pipeline over the 32 images ---
    for (int i = 0; i < NIMG; ++i) {
        const int buf = i & 1;
        if (wave == 0) {                      // scalar branch: one DMA issuer
            if (i + 1 < NIMG) {
                tdm_load_tile(img + (size_t)(i + 1) * W * H
                                  + (size_t)by0 * W + bx0,
                              base_lds + (unsigned)((buf ^ 1) * BUFE * 4),
                              bw, bh);
                __builtin_amdgcn_s_wait_tensorcnt(1); // oldest (tile i) done
            } else {
                __builtin_amdgcn_s_wait_tensorcnt(0);
            }
        }
        __syncthreads();                      // publish tile i to all waves

        const float* sb = smem + buf * BUFE;
        float* op = out + ((size_t)i * NANG + ang) * (size_t)W * H;
#pragma unroll
        for (int g = 0; g < 4; ++g) {
            const int ly = (lane >> 4) + 16 * g;
            float r0, r1, r2, r3;
            float* rp[4] = {&r0, &r1, &r2, &r3};
#pragma unroll
            for (int v = 0; v < 4; ++v) {
                const int idx = g * 4 + v;
                float v00 = sb[o00[idx]], v01 = sb[o01[idx]];
                float v10 = sb[o10[idx]], v11 = sb[o11[idx]];
                *rp[v] = ay0[idx] * (ax0[idx] * v00 + ax1[idx] * v01)
                       + ay1[idx] * (ax0[idx] * v10 + ax1[idx] * v11);
            }
            v4f val = {r0, r1, r2, r3};
            float* dst = op + (size_t)(ty0 + ly) * W + tx0 + (lane & 15) * 4;
            __builtin_nontemporal_store(val, (v4f*)dst); // NT B128 store
        }
        __syncthreads();   // everyone done reading buf before it is reloaded
    }
#endif
}

extern "C" void kernel_launch(void* const* d_in, const int* in_sizes, int n_in,
                              void* d_out, int out_size, void* d_ws,
                              size_t ws_size, hipStream_t stream) {
    (void)in_sizes; (void)n_in; (void)out_size; (void)d_ws; (void)ws_size;
    const float* img    = (const float*)d_in[0];  // (32,1,512,512) f32
    const float* angles = (const float*)d_in[1];  // (8,) f32
    float* out          = (float*)d_out;          // (32,8,512,512) f32

    dim3 grid(W / TS, H / TS, NANG);              // 8 x 8 x 8 workgroups
    dim3 block(256);                              // 8 waves (wave32)
    rotate_batch_kernel<<<grid, block, 0, stream>>>(img, angles, out);
}